// Mamba_Layer_39359080301085
// MI455X (gfx1250) — compile-verified
//
#include <hip/hip_runtime.h>
#include <hip/hip_bf16.h>
#include <cstdint>

typedef float v2f __attribute__((ext_vector_type(2)));
typedef float v8f __attribute__((ext_vector_type(8)));

#define B_      4
#define L_      2048
#define D_MODEL 1024
#define D_IN    2048
#define D_STATE 16
#define D_CONV  4
#define DT_RANK 64
#define LN_EPS  1e-5f
#define BL      (B_ * L_)   // 8192 rows

// gfx1250 async global->LDS staging, if this toolchain declares the builtins.
// Probe round 2: builtin exists; param 1 is 'int __attribute__((vector_size(16)))
// __device__ *' (global-AS int4*), param 2 is the LDS-side pointer.
#if __has_builtin(__builtin_amdgcn_global_load_async_to_lds_b128) && \
    __has_builtin(__builtin_amdgcn_s_wait_asynccnt)
#define USE_ASYNC_LDS 1
typedef int v4i __attribute__((vector_size(4 * sizeof(int))));
typedef __attribute__((address_space(1))) v4i g_v4i;   // global
typedef __attribute__((address_space(3))) v4i l_v4i;   // LDS
#else
#define USE_ASYNC_LDS 0
#endif

// ---------------------------------------------------------------------------
// Generic fp32 WMMA GEMM:  C[M,N] = A[M,K] @ B[K,N]   (row-major, fp32 exact)
// Requirements: M % 128 == 0, K % 32 == 0, lda/ldb % 4 == 0. N arbitrary.
// Workgroup = 128 threads (4 waves). Tile 128(M) x 64(N); each wave computes
// a 32x64 strip (2 m-substrips x 4 n-subtiles, 8 x v8f accumulators).
// K staged in chunks of 32 through LDS:
//   As[128][36]     : A row-major, stride 36 (banks of rows k vs k+2 disjoint)
//   Bsp[16][80][2]  : B K-pair interleaved -> fragment = one aligned 8B read;
//                     row stride 160 dwords == 32 mod 64 -> lane-halves hit
//                     disjoint bank sets.
// ---------------------------------------------------------------------------
__global__ __launch_bounds__(128) void gemm_f32_wmma(
    const float* __restrict__ A, const float* __restrict__ B,
    float* __restrict__ C, int M, int N, int K, int lda, int ldb, int ldc)
{
  __shared__ __align__(16) float As[128][36];
  __shared__ __align__(16) float Bsp[16][80][2];

  const int tid   = threadIdx.x;
  const int wave  = tid >> 5;
  const int lane  = tid & 31;
  const int mTile = blockIdx.y * 128;
  const int nTile = blockIdx.x * 64;
  const int lidx  = lane & 15;          // M index (A-frag) / N index (B,C-frag)
  const int kHalf = (lane >> 4) << 1;   // K base: lanes 0-15 -> 0, 16-31 -> 2
  const bool fullN = (nTile + 64 <= N);

  v8f acc[2][4] = {};

  for (int k0 = 0; k0 < K; k0 += 32) {
    // ---- stage A tile: 128 rows x 32 cols (1024 float4, 8 per thread) ----
#pragma unroll
    for (int i = 0; i < 8; ++i) {
      int lin = tid + i * 128;          // float4 index 0..1023
      int r = lin >> 3;
      int c = (lin & 7) << 2;
      const float* src = A + (size_t)(mTile + r) * lda + (size_t)(k0 + c);
#if USE_ASYNC_LDS
      __builtin_amdgcn_global_load_async_to_lds_b128(
          (g_v4i*)src, (l_v4i*)&As[r][c], 0, 0);
#else
      const float4 v = *reinterpret_cast<const float4*>(src);
      As[r][c + 0] = v.x; As[r][c + 1] = v.y;
      As[r][c + 2] = v.z; As[r][c + 3] = v.w;
#endif
    }
    // ---- stage B tile: 32 rows x 64 cols into K-pair-interleaved layout ----
    if (fullN) {
#pragma unroll
      for (int i = 0; i < 4; ++i) {
        int lin = tid + i * 128;
        int r = lin >> 4;
        int c = (lin & 15) << 2;
        const float4 v = *reinterpret_cast<const float4*>(
            B + (size_t)(k0 + r) * ldb + (size_t)(nTile + c));
        float* dst = &Bsp[r >> 1][c][r & 1];  // consecutive c -> stride 2 floats
        dst[0] = v.x; dst[2] = v.y; dst[4] = v.z; dst[6] = v.w;
      }
    } else {                            // ragged N (e.g. N = 96): scalar+guard
#pragma unroll
      for (int i = 0; i < 16; ++i) {
        int lin = tid + i * 128;        // 0..2047
        int r = lin >> 6;
        int c = lin & 63;
        Bsp[r >> 1][c][r & 1] =
            (nTile + c < N) ? B[(size_t)(k0 + r) * ldb + (size_t)(nTile + c)]
                            : 0.0f;
      }
    }
    // prefetch next K-chunk of A into the cache hierarchy
    if (k0 + 32 < K) {
      int pr = tid >> 1;
      int pc = (tid & 1) << 4;
      __builtin_prefetch(A + (size_t)(mTile + 2 * pr) * lda + (size_t)(k0 + 32 + pc), 0, 3);
    }
#if USE_ASYNC_LDS
    __builtin_amdgcn_s_wait_asynccnt(0);
#endif
    __syncthreads();

    // ---- 8 K-steps of 4; 2 A-frag + 4 B-frag loads feed 8 WMMAs ----
#pragma unroll
    for (int kk = 0; kk < 32; kk += 4) {
      const v2f a0 = *reinterpret_cast<const v2f*>(&As[wave * 32 + lidx][kk + kHalf]);
      const v2f a1 = *reinterpret_cast<const v2f*>(&As[wave * 32 + 16 + lidx][kk + kHalf]);
      const int kp = (kk + kHalf) >> 1;
#pragma unroll
      for (int nt = 0; nt < 4; ++nt) {
        const v2f b = *reinterpret_cast<const v2f*>(&Bsp[kp][nt * 16 + lidx][0]);
        acc[0][nt] = __builtin_amdgcn_wmma_f32_16x16x4_f32(
            false, a0, false, b, (short)0, acc[0][nt], false, false);
        acc[1][nt] = __builtin_amdgcn_wmma_f32_16x16x4_f32(
            false, a1, false, b, (short)0, acc[1][nt], false, false);
      }
    }
    __syncthreads();
  }

  // ---- store D: VGPR r holds row (r + 8*(lane>=16)) of each 16x16 tile ----
  const int rowHi = (lane >> 4) << 3;
#pragma unroll
  for (int ms = 0; ms < 2; ++ms) {
#pragma unroll
    for (int nt = 0; nt < 4; ++nt) {
      int gn = nTile + nt * 16 + lidx;
      if (gn < N) {
#pragma unroll
        for (int r = 0; r < 8; ++r) {
          int gm = mTile + wave * 32 + ms * 16 + rowHi + r;
          C[(size_t)gm * ldc + gn] = acc[ms][nt][r];
        }
      }
    }
  }
}

// ---------------------------------------------------------------------------
// Depthwise causal conv (D_CONV=4) + bias + SiLU.  xz holds [x_in | z].
// ---------------------------------------------------------------------------
__global__ __launch_bounds__(256) void conv_silu_kernel(
    const float* __restrict__ xz, const float* __restrict__ conv_w,
    const float* __restrict__ conv_b, float* __restrict__ xc)
{
  size_t idx = (size_t)blockIdx.x * 256 + threadIdx.x;   // over B*L*D_IN
  if (idx >= (size_t)BL * D_IN) return;
  int d = (int)(idx % D_IN);
  int t = (int)((idx / D_IN) % L_);
  int b = (int)(idx / ((size_t)D_IN * L_));
  float acc = conv_b[d];
#pragma unroll
  for (int k = 0; k < D_CONV; ++k) {
    int tt = t + k - (D_CONV - 1);
    if (tt >= 0)
      acc += xz[(size_t)(b * L_ + tt) * (2 * D_IN) + d] * conv_w[d * D_CONV + k];
  }
  float s = acc / (1.0f + __expf(-acc));                 // SiLU
  xc[(size_t)(b * L_ + t) * D_IN + d] = s;
}

// ---------------------------------------------------------------------------
// delta = softplus(delta_pre + b_dt[d])   (in place)
// ---------------------------------------------------------------------------
__global__ __launch_bounds__(256) void softplus_bias_kernel(
    float* __restrict__ delta, const float* __restrict__ b_dt, size_t n)
{
  size_t i = (size_t)blockIdx.x * 256 + threadIdx.x;
  if (i >= n) return;
  float v = delta[i] + b_dt[i % D_IN];
  delta[i] = (v > 20.0f) ? v : log1pf(__expf(v));
}

// ---------------------------------------------------------------------------
// Selective scan. One thread per (b, d) channel; 16 states live in VGPRs.
// B_t / C_t (shared by all channels of a batch) staged via LDS in 64-step
// chunks -> broadcast reads. SiLU(z) gating + skip (u*D) fused at the end.
// ---------------------------------------------------------------------------
__global__ __launch_bounds__(256) void scan_kernel(
    const float* __restrict__ xz, const float* __restrict__ xc,
    const float* __restrict__ dbc, const float* __restrict__ delta,
    const float* __restrict__ A_log, const float* __restrict__ Dp,
    float* __restrict__ y)
{
  __shared__ float sB[64][D_STATE];
  __shared__ float sC[64][D_STATE];
  const int tid = threadIdx.x;
  const int b   = blockIdx.y;
  const int d   = blockIdx.x * 256 + tid;

  float Aa[D_STATE];
#pragma unroll
  for (int s = 0; s < D_STATE; ++s) Aa[s] = -__expf(A_log[d * D_STATE + s]);
  const float Dd = Dp[d];
  float h[D_STATE] = {};

  for (int t0 = 0; t0 < L_; t0 += 64) {
    // stage B/C for 64 timesteps: 64 * 32 floats, 8 per thread
#pragma unroll
    for (int i = 0; i < 8; ++i) {
      int lin = tid + i * 256;          // 0..2047
      int tt  = lin >> 5;
      int j   = lin & 31;
      float v = dbc[(size_t)(b * L_ + t0 + tt) * 96 + DT_RANK + j];
      if (j < D_STATE) sB[tt][j] = v;
      else             sC[tt][j - D_STATE] = v;
    }
    __syncthreads();

    for (int tt = 0; tt < 64; ++tt) {
      size_t row  = (size_t)(b * L_) + t0 + tt;
      size_t base = row * D_IN + d;
      float u  = xc[base];
      float dl = delta[base];
      float zv = xz[row * (2 * D_IN) + D_IN + d];
      float du = dl * u;
      float yv = 0.0f;
#pragma unroll
      for (int s = 0; s < D_STATE; ++s) {
        float dA = __expf(dl * Aa[s]);
        h[s] = h[s] * dA + du * sB[tt][s];
        yv  += h[s] * sC[tt][s];
      }
      yv += u * Dd;
      float sz = zv / (1.0f + __expf(-zv));
      y[base] = yv * sz;
    }
    __syncthreads();
  }
}

// ---------------------------------------------------------------------------
// out = LayerNorm(x + yo) * gamma + beta.  One block per row of 1024.
// ---------------------------------------------------------------------------
__global__ __launch_bounds__(256) void residual_ln_kernel(
    const float* __restrict__ x, const float* __restrict__ yo,
    const float* __restrict__ gamma, const float* __restrict__ beta,
    float* __restrict__ out)
{
  __shared__ float red[256];
  const int row = blockIdx.x;
  const int tid = threadIdx.x;

  float hv[4];
  float sum = 0.0f;
#pragma unroll
  for (int i = 0; i < 4; ++i) {
    int c = tid + i * 256;
    float v = x[(size_t)row * D_MODEL + c] + yo[(size_t)row * D_MODEL + c];
    hv[i] = v;
    sum += v;
  }
  red[tid] = sum;
  __syncthreads();
  for (int s = 128; s > 0; s >>= 1) {
    if (tid < s) red[tid] += red[tid + s];
    __syncthreads();
  }
  const float mu = red[0] * (1.0f / D_MODEL);
  __syncthreads();

  float vs = 0.0f;
#pragma unroll
  for (int i = 0; i < 4; ++i) {
    float dv = hv[i] - mu;
    vs += dv * dv;
  }
  red[tid] = vs;
  __syncthreads();
  for (int s = 128; s > 0; s >>= 1) {
    if (tid < s) red[tid] += red[tid + s];
    __syncthreads();
  }
  const float inv = rsqrtf(red[0] * (1.0f / D_MODEL) + LN_EPS);

#pragma unroll
  for (int i = 0; i < 4; ++i) {
    int c = tid + i * 256;
    out[(size_t)row * D_MODEL + c] = (hv[i] - mu) * inv * gamma[c] + beta[c];
  }
}

// ---------------------------------------------------------------------------
extern "C" void kernel_launch(void* const* d_in, const int* in_sizes, int n_in,
                              void* d_out, int out_size, void* d_ws, size_t ws_size,
                              hipStream_t stream)
{
  (void)in_sizes; (void)n_in; (void)out_size; (void)ws_size;

  const float* x      = (const float*)d_in[0];
  const float* W_in   = (const float*)d_in[1];
  const float* conv_w = (const float*)d_in[2];
  const float* conv_b = (const float*)d_in[3];
  const float* W_x    = (const float*)d_in[4];
  const float* W_dt   = (const float*)d_in[5];
  const float* b_dt   = (const float*)d_in[6];
  const float* A_log  = (const float*)d_in[7];
  const float* Dp     = (const float*)d_in[8];
  const float* W_out  = (const float*)d_in[9];
  const float* gamma  = (const float*)d_in[10];
  const float* beta   = (const float*)d_in[11];
  float* out = (float*)d_out;

  float* ws = (float*)d_ws;
  size_t off = 0;
  float* xz    = ws + off; off += (size_t)BL * 2 * D_IN;   // 8192 x 4096
  float* xc    = ws + off; off += (size_t)BL * D_IN;       // 8192 x 2048
  float* dbc   = ws + off; off += (size_t)BL * 96;         // 8192 x 96
  float* delta = ws + off; off += (size_t)BL * D_IN;       // 8192 x 2048
  float* yb    = ws + off; off += (size_t)BL * D_IN;       // 8192 x 2048
  float* yo    = ws + off; off += (size_t)BL * D_MODEL;    // 8192 x 1024

  // 1) xz = x @ W_in          M=8192 N=4096 K=1024
  gemm_f32_wmma<<<dim3(4096 / 64, BL / 128), 128, 0, stream>>>(
      x, W_in, xz, BL, 4096, 1024, 1024, 4096, 4096);

  // 2) xc = SiLU(causal_conv(x_in) + b)
  conv_silu_kernel<<<(BL * D_IN) / 256, 256, 0, stream>>>(xz, conv_w, conv_b, xc);

  // 3) dbc = xc @ W_x         M=8192 N=96 K=2048
  gemm_f32_wmma<<<dim3(2, BL / 128), 128, 0, stream>>>(
      xc, W_x, dbc, BL, 96, 2048, 2048, 96, 96);

  // 4) delta = softplus(dt @ W_dt + b_dt)   (dt = dbc[:, :64], lda=96)
  gemm_f32_wmma<<<dim3(2048 / 64, BL / 128), 128, 0, stream>>>(
      dbc, W_dt, delta, BL, 2048, 64, 96, 2048, 2048);
  softplus_bias_kernel<<<(BL * D_IN) / 256, 256, 0, stream>>>(
      delta, b_dt, (size_t)BL * D_IN);

  // 5) selective scan -> yb (includes +u*D and *SiLU(z))
  scan_kernel<<<dim3(D_IN / 256, B_), 256, 0, stream>>>(
      xz, xc, dbc, delta, A_log, Dp, yb);

  // 6) yo = yb @ W_out        M=8192 N=1024 K=2048
  gemm_f32_wmma<<<dim3(1024 / 64, BL / 128), 128, 0, stream>>>(
      yb, W_out, yo, BL, 1024, 2048, 2048, 1024, 1024);

  // 7) out = LayerNorm(x + yo)
  residual_ln_kernel<<<BL, 256, 0, stream>>>(x, yo, gamma, beta, out);
}